// Phngb_81973745811696
// MI455X (gfx1250) — compile-verified
//
#include <hip/hip_runtime.h>
#include <hip/hip_bf16.h>

typedef __attribute__((ext_vector_type(2))) float v2f;
typedef __attribute__((ext_vector_type(8))) float v8f;

#define NPTS   4096   // number of points (columns of coordinates)
#define KDIM   256    // dimensionality of each point (rows of coordinates)
#define NNB    16     // neighbors
#define CHUNK  512    // columns of the distance matrix processed per LDS pass

// ---------------------------------------------------------------------------
// Kernel 1: squared norms of each column of coordinates (256 x 4096).
// Threads j are adjacent -> coords[k*4096 + j] loads are fully coalesced.
// ---------------------------------------------------------------------------
__global__ __launch_bounds__(256) void knn_sqnorm_kernel(
    const float* __restrict__ coords, float* __restrict__ sq) {
  int j = blockIdx.x * blockDim.x + threadIdx.x;  // 0..4095
  float s = 0.0f;
  for (int k = 0; k < KDIM; ++k) {
    float v = coords[k * NPTS + j];
    s += v * v;
  }
  sq[j] = s;
}

// ---------------------------------------------------------------------------
// Kernel 2: fused Gram-matrix (f32 WMMA) + streaming top-16 per row.
// One workgroup (8 wave32) owns 16 rows of the 4096x4096 distance matrix.
// For each 512-column chunk: 8 waves x 4 tiles of 16 cols, each tile
// accumulated with 64 steps of v_wmma_f32_16x16x4_f32 over K=256.
// Distances live only in LDS; a sorted top-16 carry is merged per chunk.
// ---------------------------------------------------------------------------
__global__ __launch_bounds__(256) void knn_topk_kernel(
    const float* __restrict__ coords, const float* __restrict__ sq,
    int* __restrict__ flatIdx) {
  __shared__ float tile[16][CHUNK];   // 32 KB distance chunk
  __shared__ float carryV[16][NNB];   // running top-16 values (ascending)
  __shared__ int   carryI[16][NNB];   // running top-16 indices
  __shared__ float newV[16][NNB];
  __shared__ int   newI[16][NNB];

  const int tid  = threadIdx.x;
  const int lane = tid & 31;
  const int w    = tid >> 5;        // wave id 0..7
  const int half = lane >> 4;       // 0: lanes 0-15, 1: lanes 16-31
  const int n    = lane & 15;
  const int i0   = blockIdx.x * 16; // first distance-row of this block

  // init carry
  {
    int r = tid >> 4, t = tid & 15;
    carryV[r][t] = __builtin_inff();
    carryI[r][t] = 0x7FFFFFFF;
  }
  // sq for this block's rows (per-lane: rows half*8 .. half*8+7)
  float sqRow[8];
#pragma unroll
  for (int r = 0; r < 8; ++r) sqRow[r] = sq[i0 + half * 8 + r];
  const int colA = i0 + n;  // A-matrix column (point index) for this lane
  __syncthreads();

  for (int chunk = 0; chunk < NPTS / CHUNK; ++chunk) {
    const int cbase = chunk * CHUNK;
    int jb[4];
#pragma unroll
    for (int u = 0; u < 4; ++u) jb[u] = cbase + w * 64 + u * 16 + n;

    v8f acc[4];
#pragma unroll
    for (int u = 0; u < 4; ++u) acc[u] = v8f{0.f,0.f,0.f,0.f,0.f,0.f,0.f,0.f};

    // K loop: 64 steps of 16x16x4 f32 WMMA. Per ISA layout for 32-bit A/B:
    // VGPR0 holds K=k (lanes 0-15) / K=k+2 (lanes 16-31), VGPR1 K=k+1/k+3.
    for (int t = 0; t < KDIM / 4; ++t) {
      const float* ck = coords + (size_t)((t << 2) + (half << 1)) * NPTS;
      v2f a;
      a.x = ck[colA];
      a.y = ck[colA + NPTS];
#pragma unroll
      for (int u = 0; u < 4; ++u) {
        v2f b;
        b.x = ck[jb[u]];
        b.y = ck[jb[u] + NPTS];
        acc[u] = __builtin_amdgcn_wmma_f32_16x16x4_f32(
            false, a, false, b, (short)0, acc[u], false, false);
      }
    }

    // Convert Gram tile -> clamped squared distance, write to LDS.
    // C/D layout: VGPR r = row r (lanes 0-15) / row r+8 (lanes 16-31).
#pragma unroll
    for (int u = 0; u < 4; ++u) {
      float sqCol = sq[jb[u]];
      int p = w * 64 + u * 16 + n;
#pragma unroll
      for (int r = 0; r < 8; ++r) {
        float d = fmaxf(sqRow[r] + sqCol - 2.0f * acc[u][r], 0.0f);
        tile[half * 8 + r][p] = d;
      }
    }
    __syncthreads();

    // Streaming top-16 merge: wave w handles rows 2w and 2w+1.
    for (int rsel = 0; rsel < 2; ++rsel) {
      const int r = 2 * w + rsel;
      for (int round = 0; round < NNB; ++round) {
        float bv = __builtin_inff();
        int bi = 0x7FFFFFFF, bp = -1;
        for (int p = lane; p < CHUNK + NNB; p += 32) {
          float v; int gi;
          if (p < CHUNK) { v = tile[r][p];          gi = cbase + p; }
          else           { v = carryV[r][p - CHUNK]; gi = carryI[r][p - CHUNK]; }
          if (v < bv || (v == bv && gi < bi)) { bv = v; bi = gi; bp = p; }
        }
#pragma unroll
        for (int off = 16; off >= 1; off >>= 1) {
          float ov = __shfl_xor(bv, off, 32);
          int   oi = __shfl_xor(bi, off, 32);
          int   op = __shfl_xor(bp, off, 32);
          if (ov < bv || (ov == bv && oi < bi)) { bv = ov; bi = oi; bp = op; }
        }
        if (lane == 0) {
          newV[r][round] = bv;
          newI[r][round] = bi;
          if (bp < CHUNK) tile[r][bp] = __builtin_inff();
          else            carryV[r][bp - CHUNK] = __builtin_inff();
        }
        __builtin_amdgcn_wave_barrier();  // order LDS remove before next scan
      }
      if (lane < NNB) {
        carryV[r][lane] = newV[r][lane];
        carryI[r][lane] = newI[r][lane];
      }
      __builtin_amdgcn_wave_barrier();
    }
    __syncthreads();  // protect tile before next chunk's WMMA writes
  }

  // Emit: lanes 0-15 -> row 2w, lanes 16-31 -> row 2w+1 (ascending distance)
  {
    int r = 2 * w + half;
    flatIdx[(i0 + r) * NNB + n] = carryI[r][n];
  }
}

// ---------------------------------------------------------------------------
// Kernel 3: gather. out = [ xs[:, flat_idx] (1024x65536) ;
//                           coords[:, flat_idx] (256x65536) ] flat.
// float4 groups: contiguous 128-bit stores, random reads within one cached
// 16KB source row. This is the HBM-bound stage (~335 MB written).
// ---------------------------------------------------------------------------
__global__ __launch_bounds__(256) void knn_gather_kernel(
    const float* __restrict__ xs, const float* __restrict__ coords,
    const int* __restrict__ flatIdx, float* __restrict__ out) {
  const long long FLAT   = (long long)NPTS * NNB;       // 65536
  const long long GROUPS = FLAT / 4;                    // 16384 float4 / row
  const long long XS_G   = 1024LL * GROUPS;
  long long g = (long long)blockIdx.x * blockDim.x + threadIdx.x;

  const float* src;
  float* dst;
  long long f4;
  if (g < XS_G) {
    long long row = g / GROUPS;
    f4 = g - row * GROUPS;
    src = xs + row * (long long)NPTS;
    dst = out + row * FLAT;
  } else {
    long long h = g - XS_G;
    long long row = h / GROUPS;
    f4 = h - row * GROUPS;
    src = coords + row * (long long)NPTS;
    dst = out + 1024LL * FLAT + row * FLAT;
  }
  int4 id = reinterpret_cast<const int4*>(flatIdx)[f4];
  float4 o;
  o.x = src[id.x];
  o.y = src[id.y];
  o.z = src[id.z];
  o.w = src[id.w];
  reinterpret_cast<float4*>(dst)[f4] = o;
}

// ---------------------------------------------------------------------------
extern "C" void kernel_launch(void* const* d_in, const int* in_sizes, int n_in,
                              void* d_out, int out_size, void* d_ws,
                              size_t ws_size, hipStream_t stream) {
  const float* xs     = (const float*)d_in[0];  // (1024, 4096)
  const float* coords = (const float*)d_in[1];  // (256, 4096)
  // d_in[2]: nb_neighbors == 16 (compile-time NNB)

  float* sq      = (float*)d_ws;                              // 4096 floats
  int*   flatIdx = (int*)((char*)d_ws + NPTS * sizeof(float)); // 65536 ints
  float* out     = (float*)d_out;

  knn_sqnorm_kernel<<<NPTS / 256, 256, 0, stream>>>(coords, sq);
  knn_topk_kernel<<<NPTS / 16, 256, 0, stream>>>(coords, sq, flatIdx);

  long long groups = (1024LL + 256LL) * ((long long)NPTS * NNB / 4);
  knn_gather_kernel<<<(int)(groups / 256), 256, 0, stream>>>(xs, coords,
                                                             flatIdx, out);
}